// Model_1468878815301
// MI455X (gfx1250) — compile-verified
//
#include <hip/hip_runtime.h>

typedef __attribute__((ext_vector_type(2))) float v2f;
typedef __attribute__((ext_vector_type(4))) float v4f;
typedef __attribute__((ext_vector_type(8))) float v8f;

#define WAVES_PER_BLOCK 4
#define BLOCK_THREADS   (WAVES_PER_BLOCK * 32)
#define ACT_STRIDE      68                      // floats; 272B, 16B aligned, bank-spread
#define ACT_FLOATS      (ACT_STRIDE * 32)       // 2176 floats per buffer per wave
#define Q_STRIDE        20                      // floats; 80B, 16B aligned
#define Q_FLOATS        (Q_STRIDE * 32)         // 640 floats per wave
#define WAVE_LDS_FLOATS (2 * ACT_FLOATS + Q_FLOATS)  // 4992 floats = 19968 B per wave

__device__ __forceinline__ v8f wmma_f32_16x16x4(v2f a, v2f b, v8f c) {
  // 8-arg VOP3P form: (neg_a, A, neg_b, B, c_mod, C, reuse_a, reuse_b)
  return __builtin_amdgcn_wmma_f32_16x16x4_f32(false, a, false, b, (short)0, c,
                                               false, false);
}

__device__ __forceinline__ void lds_fence() {
  // Same-wave cross-lane LDS visibility: order DS stores before subsequent loads.
  __asm__ volatile("s_wait_dscnt 0x0" ::: "memory");
}

// One dense layer: out(M x 16batch) = relu( W(M x K) * actIn(K x 16batch) + bias ).
// A/B f32 WMMA operand layout: low lanes carry K-pair {4s,4s+1}, high lanes {4s+2,4s+3}.
template <int K, int M, int WROWS, int OUT_STRIDE>
__device__ __forceinline__ void mlp_layer(const float* __restrict__ W,
                                          const float* __restrict__ bias,
                                          const float* actIn, float* actOut,
                                          int row, int hi, int elemCol) {
  lds_fence();
#pragma unroll
  for (int mt = 0; mt < M / 16; ++mt) {
    const int Mbase = mt * 16;
    int wr = Mbase + row;                 // weight row this lane feeds (A matrix M index)
    if (wr > WROWS - 1) wr = WROWS - 1;   // clamp for padded final layer (rows 12..15 unused)
    v8f acc = {};
#pragma unroll
    for (int s = 0; s < K / 4; ++s) {
      v2f a = *(const v2f*)(W + wr * K + 4 * s + hi * 2);
      v2f b = *(const v2f*)(actIn + elemCol * ACT_STRIDE + 4 * s + hi * 2);
      acc = wmma_f32_16x16x4(a, b, acc);
    }
    // D layout: VGPR r holds feature Mbase + r (low lanes) / Mbase + 8 + r (high lanes),
    // column N == lane&15 == elem column. Apply bias + relu, stage [elem][feature].
    float* dst = actOut + elemCol * OUT_STRIDE + Mbase + hi * 8;
    v4f lo = {}, hi4 = {};
#pragma unroll
    for (int r = 0; r < 8; ++r) {
      int f = Mbase + hi * 8 + r;
      if (f > WROWS - 1) f = WROWS - 1;
      float v = acc[r] + bias[f];
      v = v > 0.0f ? v : 0.0f;
      if (r < 4) lo[r] = v; else hi4[r - 4] = v;
    }
    *(v4f*)(dst)     = lo;
    *(v4f*)(dst + 4) = hi4;
  }
}

__global__ __launch_bounds__(BLOCK_THREADS) void fk_mlp_kernel(
    const float* __restrict__ mc,
    const float* __restrict__ W1, const float* __restrict__ b1,
    const float* __restrict__ W2, const float* __restrict__ b2,
    const float* __restrict__ W3, const float* __restrict__ b3,
    const float* __restrict__ W4, const float* __restrict__ b4,
    const float* __restrict__ W5, const float* __restrict__ b5,
    const float* __restrict__ twist,
    const float* __restrict__ init_p, const float* __restrict__ init_rpy,
    float* __restrict__ out, int batch) {
  __shared__ float lds[WAVES_PER_BLOCK * WAVE_LDS_FLOATS];

  const int tid  = threadIdx.x;
  const int wave = tid >> 5;
  const int lane = tid & 31;
  const int row  = lane & 15;   // M index (A) / batch column (B,D)
  const int hi   = lane >> 4;   // which K-pair of the 4-wide K slice

  float* actA = lds + wave * WAVE_LDS_FLOATS;
  float* actB = actA + ACT_FLOATS;
  float* qbuf = actB + ACT_FLOATS;

  const long elem_base = (long)(blockIdx.x * WAVES_PER_BLOCK + wave) * 32;

  // ---------------- MLP: two 16-column batch tiles per wave ----------------
#pragma unroll
  for (int t = 0; t < 2; ++t) {
    const int elemCol = t * 16 + row;
    long elemG = elem_base + elemCol;
    if (elemG > batch - 1) elemG = batch - 1;   // keep EXEC full for WMMA

    // Layer 1 (4 -> 16): B straight from global (motor_control row-major, K=4).
    {
      v2f a = *(const v2f*)(W1 + row * 4 + hi * 2);
      v2f b = *(const v2f*)(mc + elemG * 4 + hi * 2);
      v8f acc = {};
      acc = wmma_f32_16x16x4(a, b, acc);
      float* dst = actA + elemCol * ACT_STRIDE + hi * 8;
      v4f lo = {}, hi4 = {};
#pragma unroll
      for (int r = 0; r < 8; ++r) {
        float v = acc[r] + b1[hi * 8 + r];
        v = v > 0.0f ? v : 0.0f;
        if (r < 4) lo[r] = v; else hi4[r - 4] = v;
      }
      *(v4f*)(dst)     = lo;
      *(v4f*)(dst + 4) = hi4;
    }
    mlp_layer<16, 32, 32, ACT_STRIDE>(W2, b2, actA, actB, row, hi, elemCol);
    mlp_layer<32, 64, 64, ACT_STRIDE>(W3, b3, actB, actA, row, hi, elemCol);
    mlp_layer<64, 32, 32, ACT_STRIDE>(W4, b4, actA, actB, row, hi, elemCol);
    mlp_layer<32, 16, 12, Q_STRIDE  >(W5, b5, actB, qbuf, row, hi, elemCol);
  }
  lds_fence();

  // ---------------- FK: one batch element per lane ----------------
  float qv[12];
  {
    const v4f* qp = (const v4f*)(qbuf + lane * Q_STRIDE);
    v4f q0 = qp[0], q1 = qp[1], q2 = qp[2];
    qv[0] = q0[0]; qv[1] = q0[1]; qv[2]  = q0[2]; qv[3]  = q0[3];
    qv[4] = q1[0]; qv[5] = q1[1]; qv[6]  = q1[2]; qv[7]  = q1[3];
    qv[8] = q2[0]; qv[9] = q2[1]; qv[10] = q2[2]; qv[11] = q2[3];
  }

  // T kept as affine 3x4 (bottom row is always [0,0,0,1]).
  float T00 = 1.f, T01 = 0.f, T02 = 0.f, T03 = 0.f;
  float T10 = 0.f, T11 = 1.f, T12 = 0.f, T13 = 0.f;
  float T20 = 0.f, T21 = 0.f, T22 = 1.f, T23 = 0.f;

#pragma unroll
  for (int j = 0; j < 12; ++j) {
    // Wave-uniform twist constants (scalar loads / computed once per wave).
    const float rx = twist[j * 6 + 0], ry = twist[j * 6 + 1], rz = twist[j * 6 + 2];
    const float wx = twist[j * 6 + 3], wy = twist[j * 6 + 4], wz = twist[j * 6 + 5];
    const float wn  = __builtin_sqrtf(wx * wx + wy * wy + wz * wz + 1e-12f);
    const float inv = 1.0f / wn;
    const float ux = wx * inv, uy = wy * inv, uz = wz * inv;
    const float rux = rx * inv, ruy = ry * inv, ruz = rz * inv;
    // K^2 (symmetric), K*ru, K^2*ru — all uniform.
    const float kxx = -(uy * uy + uz * uz);
    const float kyy = -(ux * ux + uz * uz);
    const float kzz = -(ux * ux + uy * uy);
    const float kxy = ux * uy, kxz = ux * uz, kyz = uy * uz;
    const float krx = uy * ruz - uz * ruy;
    const float kry = uz * rux - ux * ruz;
    const float krz = ux * ruy - uy * rux;
    const float k2rx = kxx * rux + kxy * ruy + kxz * ruz;
    const float k2ry = kxy * rux + kyy * ruy + kyz * ruz;
    const float k2rz = kxz * rux + kyz * ruy + kzz * ruz;

    // Per-lane part.
    const float th = qv[j] * wn;
    const float sn = __sinf(th);
    const float cs = __cosf(th);
    const float c1 = 1.0f - cs;
    const float tm = th - sn;

    const float R00 = 1.0f + c1 * kxx;
    const float R01 = -sn * uz + c1 * kxy;
    const float R02 =  sn * uy + c1 * kxz;
    const float R10 =  sn * uz + c1 * kxy;
    const float R11 = 1.0f + c1 * kyy;
    const float R12 = -sn * ux + c1 * kyz;
    const float R20 = -sn * uy + c1 * kxz;
    const float R21 =  sn * ux + c1 * kyz;
    const float R22 = 1.0f + c1 * kzz;

    const float px = th * rux + c1 * krx + tm * k2rx;
    const float py = th * ruy + c1 * kry + tm * k2ry;
    const float pz = th * ruz + c1 * krz + tm * k2rz;

    const float n00 = T00 * R00 + T01 * R10 + T02 * R20;
    const float n01 = T00 * R01 + T01 * R11 + T02 * R21;
    const float n02 = T00 * R02 + T01 * R12 + T02 * R22;
    const float n03 = T00 * px + T01 * py + T02 * pz + T03;
    const float n10 = T10 * R00 + T11 * R10 + T12 * R20;
    const float n11 = T10 * R01 + T11 * R11 + T12 * R21;
    const float n12 = T10 * R02 + T11 * R12 + T12 * R22;
    const float n13 = T10 * px + T11 * py + T12 * pz + T13;
    const float n20 = T20 * R00 + T21 * R10 + T22 * R20;
    const float n21 = T20 * R01 + T21 * R11 + T22 * R21;
    const float n22 = T20 * R02 + T21 * R12 + T22 * R22;
    const float n23 = T20 * px + T21 * py + T22 * pz + T23;
    T00 = n00; T01 = n01; T02 = n02; T03 = n03;
    T10 = n10; T11 = n11; T12 = n12; T13 = n13;
    T20 = n20; T21 = n21; T22 = n22; T23 = n23;
  }

  // T0 = pr2t(init_p, init_rpy): uniform.
  const float rr = init_rpy[0], pp = init_rpy[1], yy = init_rpy[2];
  const float cr = __cosf(rr), sr = __sinf(rr);
  const float cp = __cosf(pp), sp = __sinf(pp);
  const float cy = __cosf(yy), sy = __sinf(yy);
  const float A00 = cy * cp, A01 = cy * sp * sr - sy * cr, A02 = cy * sp * cr + sy * sr;
  const float A10 = sy * cp, A11 = sy * sp * sr + cy * cr, A12 = sy * sp * cr - cy * sr;
  const float A20 = -sp,     A21 = cp * sr,                A22 = cp * cr;
  const float px0 = init_p[0], py0 = init_p[1], pz0 = init_p[2];

  const float o00 = T00 * A00 + T01 * A10 + T02 * A20;
  const float o01 = T00 * A01 + T01 * A11 + T02 * A21;
  const float o02 = T00 * A02 + T01 * A12 + T02 * A22;
  const float o03 = T00 * px0 + T01 * py0 + T02 * pz0 + T03;
  const float o10 = T10 * A00 + T11 * A10 + T12 * A20;
  const float o11 = T10 * A01 + T11 * A11 + T12 * A21;
  const float o12 = T10 * A02 + T11 * A12 + T12 * A22;
  const float o13 = T10 * px0 + T11 * py0 + T12 * pz0 + T13;
  const float o20 = T20 * A00 + T21 * A10 + T22 * A20;
  const float o21 = T20 * A01 + T21 * A11 + T22 * A21;
  const float o22 = T20 * A02 + T21 * A12 + T22 * A22;
  const float o23 = T20 * px0 + T21 * py0 + T22 * pz0 + T23;

  const long elemOut = elem_base + lane;
  if (elemOut < batch) {
    float* o = out + elemOut * 16;
    *(v4f*)(o + 0)  = (v4f){o00, o01, o02, o03};
    *(v4f*)(o + 4)  = (v4f){o10, o11, o12, o13};
    *(v4f*)(o + 8)  = (v4f){o20, o21, o22, o23};
    *(v4f*)(o + 12) = (v4f){0.f, 0.f, 0.f, 1.f};
  }
}

extern "C" void kernel_launch(void* const* d_in, const int* in_sizes, int n_in,
                              void* d_out, int out_size, void* d_ws, size_t ws_size,
                              hipStream_t stream) {
  const float* mc = (const float*)d_in[0];
  const float* W1 = (const float*)d_in[1];  const float* b1 = (const float*)d_in[2];
  const float* W2 = (const float*)d_in[3];  const float* b2 = (const float*)d_in[4];
  const float* W3 = (const float*)d_in[5];  const float* b3 = (const float*)d_in[6];
  const float* W4 = (const float*)d_in[7];  const float* b4 = (const float*)d_in[8];
  const float* W5 = (const float*)d_in[9];  const float* b5 = (const float*)d_in[10];
  const float* twist    = (const float*)d_in[11];
  const float* init_p   = (const float*)d_in[12];
  const float* init_rpy = (const float*)d_in[13];
  float* out = (float*)d_out;

  const int batch = in_sizes[0] / 4;                       // motor_control is (B,4)
  const int waves = (batch + 31) / 32;                     // 32 elems per wave
  const int blocks = (waves + WAVES_PER_BLOCK - 1) / WAVES_PER_BLOCK;

  fk_mlp_kernel<<<blocks, BLOCK_THREADS, 0, stream>>>(
      mc, W1, b1, W2, b2, W3, b3, W4, b4, W5, b5,
      twist, init_p, init_rpy, out, batch);
}